// ParticleCollision_4501125726537
// MI455X (gfx1250) — compile-verified
//
#include <hip/hip_runtime.h>

// ParticleCollision for MI455X (gfx1250, wave32).
// Distance matrix computed with V_WMMA_F32_16X16X4_F32:
//   A row m  = [qx, qy, qz, 1]
//   B col n  = [-2px, -2py, -2pz, pn]        (pn = |p|^2)
//   C[m][n]  = qn[m] - R^2                   (qn = |q|^2, row-constant)
//   D = A*B + C = dist2 - R^2  -> neighbor iff D <= 0.

#define NDIM     3
#define RADIUS   0.14f
#define R2       (RADIUS * RADIUS)
#define MAXGRID  96.0f
#define MAXC     128
#define B_       4
#define N_       4096
#define M_       4096
#define C_       16

typedef __attribute__((ext_vector_type(2))) float v2f;
typedef __attribute__((ext_vector_type(8))) float v8f;

// ---------------------------------------------------------------- bbox ------
__global__ void pc_bbox_kernel(const float* __restrict__ locs,
                               float* __restrict__ bbox) {
  const int b = blockIdx.x;
  const int t = threadIdx.x;                       // 256 threads
  float mn[3] = {1e30f, 1e30f, 1e30f};
  float mx[3] = {-1e30f, -1e30f, -1e30f};
  for (int i = t; i < N_; i += 256) {
    const float* p = locs + ((size_t)b * N_ + i) * 3;
    for (int d = 0; d < 3; ++d) {
      float v = p[d];
      mn[d] = fminf(mn[d], v);
      mx[d] = fmaxf(mx[d], v);
    }
  }
  __shared__ float smn[3][256], smx[3][256];
  for (int d = 0; d < 3; ++d) { smn[d][t] = mn[d]; smx[d][t] = mx[d]; }
  __syncthreads();
  for (int s = 128; s > 0; s >>= 1) {
    if (t < s) {
      for (int d = 0; d < 3; ++d) {
        smn[d][t] = fminf(smn[d][t], smn[d][t + s]);
        smx[d][t] = fmaxf(smx[d][t], smx[d][t + s]);
      }
    }
    __syncthreads();
  }
  if (t == 0) {
    for (int d = 0; d < 3; ++d) {
      bbox[b * 6 + d]     = smn[d][0];
      bbox[b * 6 + 3 + d] = smx[d][0];
    }
  }
}

// ---------------------------------------------------------------- cell id ---
__global__ void pc_cellid_kernel(const float* __restrict__ locs,
                                 const float* __restrict__ bbox,
                                 int* __restrict__ keys) {
  const int gid = blockIdx.x * blockDim.x + threadIdx.x;
  if (gid >= B_ * N_) return;
  const int b = gid / N_;
  const float* p = locs + (size_t)gid * 3;
  int ci[3], gd[3];
#pragma unroll
  for (int d = 0; d < 3; ++d) {
    const float lo = bbox[b * 6 + d];
    const float hi = bbox[b * 6 + 3 + d];
    const float g  = ceilf(fminf(fmaxf((hi - lo) / RADIUS, 0.0f), MAXGRID));
    const float center = 0.5f * (lo + hi);
    const float low2   = center - g * RADIUS * 0.5f;
    float cf = floorf((p[d] - low2) / RADIUS);
    cf = fminf(fmaxf(cf, 0.0f), g - 1.0f);
    ci[d] = (int)cf;
    gd[d] = max((int)g, 1);
  }
  keys[gid] = ci[0] * (gd[1] * gd[2]) + ci[1] * gd[2] + ci[2];
}

// ------------------------------------------------- stable rank + scatter ----
// rank_i = #{j : key_j < key_i} + #{j < i : key_j == key_i}  (stable argsort)
// Scatter locs/data/idxs into sorted order and build P = [-2x,-2y,-2z,|p|^2].
__global__ void pc_rank_kernel(const float* __restrict__ locs,
                               const float* __restrict__ data,
                               const int* __restrict__ keys,
                               float* __restrict__ locs_r,
                               float* __restrict__ data_r,
                               int* __restrict__ idxs_out,
                               float4* __restrict__ P) {
  __shared__ int sk[N_];                           // 16 KB of 320 KB LDS
  const int blocksPerBatch = N_ / 256;
  const int b     = blockIdx.x / blocksPerBatch;
  const int chunk = blockIdx.x % blocksPerBatch;
  const int t     = threadIdx.x;
  const int* kb = keys + b * N_;
  for (int i = t; i < N_; i += 256) sk[i] = kb[i];
  __syncthreads();

  const int i  = chunk * 256 + t;
  const int ki = sk[i];
  int rank = 0;
#pragma unroll 4
  for (int j = 0; j < N_; ++j) {
    const int kj = sk[j];
    rank += (kj < ki) || (kj == ki && j < i);
  }

  const float* p = locs + ((size_t)b * N_ + i) * 3;
  const float x = p[0], y = p[1], z = p[2];
  float* lr = locs_r + ((size_t)b * N_ + rank) * 3;
  lr[0] = x; lr[1] = y; lr[2] = z;

  const float4* din = (const float4*)(data + ((size_t)b * N_ + i) * C_);
  float4* dr = (float4*)(data_r + ((size_t)b * N_ + rank) * C_);
#pragma unroll
  for (int c = 0; c < C_ / 4; ++c) dr[c] = din[c];

  idxs_out[b * N_ + rank] = i;
  P[b * N_ + rank] = make_float4(-2.0f * x, -2.0f * y, -2.0f * z,
                                 x * x + y * y + z * z);
}

// ---------------------------------------------------- WMMA radius search ----
// One wave owns a 16-query tile; 4 waves per block. Walk n-tiles in ascending
// order, one V_WMMA_F32_16X16X4_F32 per 16x16 tile, ballot + prefix-popcount
// for stable in-order appends capped at MAXC, then -1 tail fill.
__global__ void __launch_bounds__(128)
pc_collide_kernel(const float* __restrict__ qlocs,
                  const float4* __restrict__ P,
                  int* __restrict__ nbr) {
  __shared__ int s_cnt[4][16];
  const int lane = threadIdx.x & 31;
  const int w    = threadIdx.x >> 5;
  const int qt   = blockIdx.x * 4 + w;             // global query tile
  const int b    = qt >> 8;                        // 256 tiles per batch
  const int rowbase = (qt & 255) * 16;
  const bool hi  = lane >= 16;
  const int  c   = lane & 15;

  if (lane < 16) s_cnt[w][lane] = 0;

  // A fragment: lanes 0-15 hold K=0,1 (qx,qy); lanes 16-31 hold K=2,3 (qz,1).
  const float* q = qlocs + ((size_t)b * M_ + rowbase + c) * 3;
  const float qx = q[0], qy = q[1], qz = q[2];
  const float qn = qx * qx + qy * qy + qz * qz;
  v2f a;
  a.x = hi ? qz : qx;
  a.y = hi ? 1.0f : qy;

  // C fragment: VGPR v = qn[row v] (lanes 0-15) / qn[row v+8] (lanes 16-31), -R^2.
  v8f cf;
#pragma unroll
  for (int v = 0; v < 8; ++v) {
    const float qlo = __shfl(qn, v, 32);
    const float qhi = __shfl(qn, v + 8, 32);
    cf[v] = (hi ? qhi : qlo) - R2;
  }

  const float4* Pb = P + (size_t)b * N_;
  int* nb = nbr + ((size_t)b * M_ + rowbase) * MAXC;

  for (int nt = 0; nt < N_ / 16; ++nt) {
    const int n0 = nt * 16 + c;                    // reordered particle index
    const float4 p = Pb[n0];
    // B fragment: VGPR0 = rows K0/K2, VGPR1 = rows K1/K3 of the 4x16 B tile.
    v2f bf;
    bf.x = hi ? p.z : p.x;
    bf.y = hi ? p.w : p.y;

    v8f d = __builtin_amdgcn_wmma_f32_16x16x4_f32(
        /*neg_a=*/false, a, /*neg_b=*/false, bf,
        /*c_mod=*/(short)0, cf, /*reuse_a=*/false, /*reuse_b=*/false);

#pragma unroll
    for (int r = 0; r < 8; ++r) {
      const bool cond = d[r] <= 0.0f;              // dist2 <= R^2
      const unsigned mask = __builtin_amdgcn_ballot_w32(cond);
      const unsigned sub  = hi ? (mask >> 16) : (mask & 0xFFFFu);
      const int row = r + (hi ? 8 : 0);
      const int cnt = s_cnt[w][row];               // wave-private, in-order DS
      const int off = __popc(sub & ((1u << c) - 1u));
      const int pos = cnt + off;
      if (cond && pos < MAXC) nb[row * MAXC + pos] = n0;
      if (c == 0) s_cnt[w][row] = cnt + __popc(sub);
    }
  }

  // Pad remaining slots with -1.
#pragma unroll 1
  for (int r = 0; r < 16; ++r) {
    const int cnt = s_cnt[w][r];
    const int start = cnt < MAXC ? cnt : MAXC;
    for (int k = lane; k < MAXC; k += 32)
      if (k >= start) nb[r * MAXC + k] = -1;
  }
}

// ---------------------------------------------------------------- launch ----
extern "C" void kernel_launch(void* const* d_in, const int* in_sizes, int n_in,
                              void* d_out, int out_size, void* d_ws, size_t ws_size,
                              hipStream_t stream) {
  const float* locs  = (const float*)d_in[0];  // [B,N,3]
  const float* data  = (const float*)d_in[1];  // [B,N,16]
  const float* qlocs = (const float*)d_in[2];  // [B,M,3]

  float* out    = (float*)d_out;
  float* locs_r = out;                                   // B*N*3
  float* data_r = out + (size_t)B_ * N_ * 3;             // B*N*16
  int*   idxs   = (int*)(out + (size_t)B_ * N_ * (3 + 16));        // B*N
  int*   nbr    = (int*)(out + (size_t)B_ * N_ * (3 + 16 + 1));    // B*M*128

  float*  bbox = (float*)d_ws;                                  // B*6 floats
  int*    keys = (int*)((char*)d_ws + 256);                     // B*N ints
  float4* P    = (float4*)((char*)d_ws + 256 +
                           (size_t)B_ * N_ * sizeof(int));      // B*N float4 (16B aligned)

  hipLaunchKernelGGL(pc_bbox_kernel, dim3(B_), dim3(256), 0, stream, locs, bbox);
  hipLaunchKernelGGL(pc_cellid_kernel, dim3((B_ * N_ + 255) / 256), dim3(256), 0,
                     stream, locs, bbox, keys);
  hipLaunchKernelGGL(pc_rank_kernel, dim3(B_ * (N_ / 256)), dim3(256), 0, stream,
                     locs, data, keys, locs_r, data_r, idxs, P);
  hipLaunchKernelGGL(pc_collide_kernel, dim3(B_ * (M_ / 16) / 4), dim3(128), 0,
                     stream, qlocs, P, nbr);
}